// modelLSTM_43112881717424
// MI455X (gfx1250) — compile-verified
//
#include <hip/hip_runtime.h>
#include <math.h>

#define SEQ   4096
#define INDIM 512
#define H     2048
#define G4    (4 * H)   // 8192 gate rows

typedef float v2f __attribute__((ext_vector_type(2)));
typedef float v4f __attribute__((ext_vector_type(4)));
typedef float v8f __attribute__((ext_vector_type(8)));

// ------------------------------------------------------------------
// prep kernels
// ------------------------------------------------------------------
// w_hh: [4H][H] row-major -> WT4: [H][H][4]  (gate-interleaved, k-major)
// WT4[(k*H + j)*4 + g] = w_hh[(g*H + j)*H + k]
__global__ void k_pack_whh(const float* __restrict__ w, float* __restrict__ wt4) {
    long i = (long)blockIdx.x * blockDim.x + threadIdx.x;
    if (i >= (long)G4 * H) return;
    int r = (int)(i / H);        // gate row 0..8191
    int k = (int)(i % H);        // source hidden index
    int g = r >> 11;             // gate 0..3 (H == 2^11)
    int j = r & (H - 1);         // hidden unit
    wt4[((long)k * H + j) * 4 + g] = w[i];
}

__global__ void k_bias(const float* __restrict__ a, const float* __restrict__ b,
                       float* __restrict__ o, int n) {
    int i = blockIdx.x * blockDim.x + threadIdx.x;
    if (i < n) o[i] = a[i] + b[i];
}

__global__ void k_zero(int* p, int n) {
    int i = blockIdx.x * blockDim.x + threadIdx.x;
    if (i < n) p[i] = 0;
}

// ------------------------------------------------------------------
// C[M,N] = A[M,K] @ W[N,K]^T  via V_WMMA_F32_16X16X4_F32
// one wave computes a 32x32 tile (4 accumulators): 1 load per WMMA.
// M,N multiples of 32, K multiple of 4.
// ------------------------------------------------------------------
__device__ __forceinline__ v8f wmma4(v2f a, v2f b, v8f c) {
    return __builtin_amdgcn_wmma_f32_16x16x4_f32(false, a, false, b,
                                                 (short)0, c, false, false);
}

__global__ void __launch_bounds__(256)
k_gemm_awt(const float* __restrict__ A, const float* __restrict__ W,
           float* __restrict__ C, int M, int N, int K) {
    const int lane = threadIdx.x & 31;
    const int wv   = threadIdx.x >> 5;            // 8 waves / block
    const long tile = (long)blockIdx.x * 8 + wv;
    const int ntile = N >> 5;                     // 32-wide tiles
    const int m0 = (int)(tile / ntile) << 5;
    const int n0 = (int)(tile % ntile) << 5;

    const int half = lane >> 4;                   // 0: K=k0,k0+1  1: K=k0+2,k0+3
    const int l    = lane & 15;
    const float* a0p = A + (long)(m0 + l) * K + 2 * half;
    const float* a1p = a0p + 16L * K;
    const float* b0p = W + (long)(n0 + l) * K + 2 * half;
    const float* b1p = b0p + 16L * K;

    v8f acc00 = {}, acc01 = {}, acc10 = {}, acc11 = {};
    for (int k0 = 0; k0 < K; k0 += 4) {
        v2f a0 = *(const v2f*)(a0p + k0);
        v2f a1 = *(const v2f*)(a1p + k0);
        v2f b0 = *(const v2f*)(b0p + k0);
        v2f b1 = *(const v2f*)(b1p + k0);
        acc00 = wmma4(a0, b0, acc00);
        acc01 = wmma4(a0, b1, acc01);
        acc10 = wmma4(a1, b0, acc10);
        acc11 = wmma4(a1, b1, acc11);
    }
    // C/D layout: VGPR r -> M = r + 8*half, N = l
    float* c00 = C + (long)m0 * N + n0 + l;
    float* c10 = c00 + 16L * N;
#pragma unroll
    for (int r = 0; r < 8; ++r) {
        const long row = (long)(r + 8 * half) * N;
        c00[row]      = acc00[r];
        c00[row + 16] = acc01[r];
        c10[row]      = acc10[r];
        c10[row + 16] = acc11[r];
    }
}

// ------------------------------------------------------------------
// persistent LSTM scan: thread j owns hidden unit j, keeps c_j in a
// register, computes its four gates from one float4 weight stream.
// one device-wide barrier per timestep (double-buffered h).
// ------------------------------------------------------------------
#define SBLK 32
#define STHR 64   // SBLK*STHR == H

__device__ __forceinline__ void grid_barrier(int* cnt, int target) {
    __builtin_amdgcn_fence(__ATOMIC_RELEASE, "agent");
    __syncthreads();
    if (threadIdx.x == 0) {
        __hip_atomic_fetch_add(cnt, 1, __ATOMIC_RELAXED, __HIP_MEMORY_SCOPE_AGENT);
        while (__hip_atomic_load(cnt, __ATOMIC_RELAXED, __HIP_MEMORY_SCOPE_AGENT) < target)
            __builtin_amdgcn_s_sleep(2);
    }
    __syncthreads();
    __builtin_amdgcn_fence(__ATOMIC_ACQUIRE, "agent");
}

__global__ void __launch_bounds__(STHR)
k_lstm_scan(const float* __restrict__ gx,     // [T][4H] input contribution
            const float* __restrict__ WT4,    // [H][H][4] gate-interleaved w_hh^T
            const float* __restrict__ bias,   // [4H] b_ih + b_hh
            float* __restrict__ hbuf,         // [2][H] double buffer
            float* __restrict__ hs,           // [T][H] or nullptr
            int T, int* cnt) {
    const int j = blockIdx.x * STHR + threadIdx.x;   // 0..H-1
    __shared__ float hl[H];
    float c = 0.f;
    float* h0 = hbuf;
    float* h1 = hbuf + H;
    h0[j] = 0.f;
    int bid = 1;
    grid_barrier(cnt, bid * SBLK); ++bid;

    const v4f* w4 = (const v4f*)WT4 + j;             // stride H in float4 units

    for (int t = 0; t < T; ++t) {
        const float* hcur = (t & 1) ? h1 : h0;
        float*       hnxt = (t & 1) ? h0 : h1;
        for (int k = threadIdx.x; k < H; k += STHR) hl[k] = hcur[k];
        __syncthreads();

        const long base = (long)t * G4;
        float ai = gx[base + j]         + bias[j];
        float af = gx[base + H + j]     + bias[H + j];
        float ag = gx[base + 2 * H + j] + bias[2 * H + j];
        float ao = gx[base + 3 * H + j] + bias[3 * H + j];

#pragma unroll 8
        for (int k = 0; k < H; ++k) {
            const float hk = hl[k];
            const v4f wv = w4[(long)k * H];          // one global_load_b128
            ai = fmaf(hk, wv.x, ai);
            af = fmaf(hk, wv.y, af);
            ag = fmaf(hk, wv.z, ag);
            ao = fmaf(hk, wv.w, ao);
        }
        const float ig = 1.f / (1.f + __expf(-ai));
        const float fg = 1.f / (1.f + __expf(-af));
        const float gg = tanhf(ag);
        const float og = 1.f / (1.f + __expf(-ao));
        c = fg * c + ig * gg;
        const float hv = og * tanhf(c);
        hnxt[j] = hv;
        if (hs) hs[(long)t * H + j] = hv;
        grid_barrier(cnt, bid * SBLK); ++bid;   // includes __syncthreads (LDS reuse safe)
    }
}

// ------------------------------------------------------------------
// heads: two dot(h, w)+b
// ------------------------------------------------------------------
__global__ void __launch_bounds__(256)
k_heads(const float* __restrict__ h,
        const float* __restrict__ wt, const float* __restrict__ bt,
        const float* __restrict__ wf, const float* __restrict__ bf,
        float* __restrict__ out) {
    __shared__ float s0[256], s1[256];
    float a = 0.f, b = 0.f;
    for (int k = threadIdx.x; k < H; k += 256) {
        const float hv = h[k];
        a = fmaf(hv, wt[k], a);
        b = fmaf(hv, wf[k], b);
    }
    s0[threadIdx.x] = a; s1[threadIdx.x] = b;
    __syncthreads();
    for (int s = 128; s > 0; s >>= 1) {
        if (threadIdx.x < s) {
            s0[threadIdx.x] += s0[threadIdx.x + s];
            s1[threadIdx.x] += s1[threadIdx.x + s];
        }
        __syncthreads();
    }
    if (threadIdx.x == 0) { out[0] = s0[0] + bt[0]; out[1] = s1[0] + bf[0]; }
}

// ------------------------------------------------------------------
extern "C" void kernel_launch(void* const* d_in, const int* in_sizes, int n_in,
                              void* d_out, int out_size, void* d_ws, size_t ws_size,
                              hipStream_t stream) {
    (void)in_sizes; (void)n_in; (void)out_size; (void)ws_size;
    const float* x     = (const float*)d_in[0];
    const float* w_ih0 = (const float*)d_in[1];
    const float* w_hh0 = (const float*)d_in[2];
    const float* b_ih0 = (const float*)d_in[3];
    const float* b_hh0 = (const float*)d_in[4];
    const float* w_ih1 = (const float*)d_in[5];
    const float* w_hh1 = (const float*)d_in[6];
    const float* b_ih1 = (const float*)d_in[7];
    const float* b_hh1 = (const float*)d_in[8];
    const float* w_t   = (const float*)d_in[9];
    const float* b_t   = (const float*)d_in[10];
    const float* w_f   = (const float*)d_in[11];
    const float* b_f   = (const float*)d_in[12];
    float* out = (float*)d_out;

    char* ws = (char*)d_ws;
    size_t off = 0;
    auto alloc = [&](size_t bytes) -> char* {
        char* p = ws + off;
        off += (bytes + 255) & ~(size_t)255;
        return p;
    };
    float* WT0   = (float*)alloc((size_t)H * G4 * 4);     // 64 MB
    float* WT1   = (float*)alloc((size_t)H * G4 * 4);     // 64 MB
    float* gx0   = (float*)alloc((size_t)SEQ * G4 * 4);   // 128 MB
    float* gx1   = (float*)alloc((size_t)SEQ * G4 * 4);   // 128 MB
    float* hs0   = (float*)alloc((size_t)SEQ * H * 4);    // 32 MB
    float* bias0 = (float*)alloc((size_t)G4 * 4);
    float* bias1 = (float*)alloc((size_t)G4 * 4);
    float* hbuf  = (float*)alloc((size_t)2 * H * 4);
    int*   cnt   = (int*)alloc(1024);                     // 2 counters, padded

    // prep: pack recurrent weights gate-interleaved k-major, fuse biases, zero barriers
    const long nt = (long)G4 * H;
    k_pack_whh<<<(int)((nt + 255) / 256), 256, 0, stream>>>(w_hh0, WT0);
    k_pack_whh<<<(int)((nt + 255) / 256), 256, 0, stream>>>(w_hh1, WT1);
    k_bias<<<(G4 + 255) / 256, 256, 0, stream>>>(b_ih0, b_hh0, bias0, G4);
    k_bias<<<(G4 + 255) / 256, 256, 0, stream>>>(b_ih1, b_hh1, bias1, G4);
    k_zero<<<1, 256, 0, stream>>>(cnt, 256);

    const int tiles  = (SEQ / 32) * (G4 / 32);             // 32768 waves
    const int blocks = tiles / 8;                          // 8 waves / block

    // layer 0: WMMA GEMM for input contribution, then persistent scan
    k_gemm_awt<<<blocks, 256, 0, stream>>>(x, w_ih0, gx0, SEQ, G4, INDIM);
    k_lstm_scan<<<SBLK, STHR, 0, stream>>>(gx0, WT0, bias0, hbuf, hs0, SEQ, cnt);

    // layer 1: WMMA GEMM on full layer-0 hidden sequence, then scan
    k_gemm_awt<<<blocks, 256, 0, stream>>>(hs0, w_ih1, gx1, SEQ, G4, H);
    k_lstm_scan<<<SBLK, STHR, 0, stream>>>(gx1, WT1, bias1, hbuf, nullptr, SEQ, cnt + 128);

    // final h (SEQ even -> buffer 0); two scalar heads
    k_heads<<<1, 256, 0, stream>>>(hbuf, w_t, b_t, w_f, b_f, out);
}